// ALCOVECell_7988639171083
// MI455X (gfx1250) — compile-verified
//
#include <hip/hip_runtime.h>
#include <math.h>

#define B_ 32
#define T_ 16
#define R_ 1024
#define D_ 64
#define U_ 64
#define EPSF 1e-6f

typedef float v2f __attribute__((ext_vector_type(2)));
typedef float v8f __attribute__((ext_vector_type(8)));

// LDS budget (floats): assoc 65536 + sim/S/d/gs/coef 5*1024 + z/att/x/gx 4*64 + red 256
#define SMEM_FLOATS (R_*U_ + 5*R_ + 4*64 + 256)
#define SMEM_BYTES  (SMEM_FLOATS * sizeof(float))   // 284,672 B < 320 KB WGP LDS

__global__ __launch_bounds__(256, 1)
void alcove_scan_kernel(const int* __restrict__ stim_set,
                        const int* __restrict__ label_idx,
                        const float* __restrict__ embed,
                        const float* __restrict__ rho_p,
                        const float* __restrict__ temp_p,
                        const float* __restrict__ lrA_p,
                        const float* __restrict__ lrW_p,
                        const float* __restrict__ beta_p,
                        float* __restrict__ out)
{
    extern __shared__ float smem[];
    float* s_assoc = smem;                  // [R_][U_] association, resident all 16 steps
    float* s_sim   = s_assoc + R_*U_;       // [R_] s[r] = exp(-beta*d)
    float* s_S     = s_sim + R_;            // [R_] S+eps (pre-root sum)
    float* s_dd    = s_S + R_;              // [R_] d = (S+eps)^(1/rho)
    float* s_gs    = s_dd + R_;             // [R_] dL/ds
    float* s_coef  = s_gs + R_;             // [R_] g_att coefficient per r
    float* s_z     = s_coef + R_;           // [64] query percept
    float* s_att   = s_z + D_;              // [64] attention (state)
    float* s_x     = s_att + D_;            // [64] class activation
    float* s_gx    = s_x + U_;              // [64] dL/dx
    float* s_red   = s_gx + U_;             // [256] reduction scratch

    const int b    = blockIdx.x;
    const int tid  = threadIdx.x;
    const int lane = tid & 31;
    const int wave = tid >> 5;

    const float rho     = rho_p[0];
    const float inv_rho = 1.0f / rho;
    const float temp    = temp_p[0];
    const float lrA     = lrA_p[0];
    const float lrW     = lrW_p[0];
    const float beta    = beta_p[0];

    // init persistent state
    for (int i = tid; i < R_*U_; i += 256) s_assoc[i] = 0.0f;
    if (tid < D_) s_att[tid] = 1.0f / (float)D_;
    __syncthreads();

    for (int t = 0; t < T_; ++t) {
        const int st  = stim_set[b*T_ + t];
        const int lab = label_idx[b*T_ + t];
        if (tid < D_) { s_z[tid] = embed[st*D_ + tid]; s_x[tid] = 0.0f; }
        __syncthreads();

        // ---- Phase A: Minkowski distance + RBF similarity (4 r per thread) ----
        for (int k = 0; k < 4; ++k) {
            const int r = tid + k*256;
            const float4* er4 = (const float4*)(embed + (size_t)r * D_);
            if (k < 3) __builtin_prefetch(embed + (size_t)(r + 256) * D_, 0, 1);
            float acc = 0.0f;
            #pragma unroll
            for (int dq = 0; dq < D_/4; ++dq) {
                float4 e = er4[dq];
                const int d0 = dq*4;
                acc += s_att[d0+0] * powf(fabsf(s_z[d0+0] - e.x) + EPSF, rho);
                acc += s_att[d0+1] * powf(fabsf(s_z[d0+1] - e.y) + EPSF, rho);
                acc += s_att[d0+2] * powf(fabsf(s_z[d0+2] - e.z) + EPSF, rho);
                acc += s_att[d0+3] * powf(fabsf(s_z[d0+3] - e.w) + EPSF, rho);
            }
            const float Sv = acc + EPSF;
            const float dv = powf(Sv, inv_rho);
            s_S[r]   = Sv;
            s_dd[r]  = dv;
            s_sim[r] = __expf(-beta * dv);
        }
        __syncthreads();

        // ---- Phase B: x[u] = sum_r s[r]*assoc[r][u]  (WMMA f32 16x16x4, M=0 row) ----
        // A fragment (similarities) hoisted across the 4 u-tiles; 4 live accumulators
        // give independent WMMA chains that cover the matrix-pipe RAW latency.
        {
            const int r_base = wave * 128;          // each wave owns 128 r's
            const int n  = lane & 15;               // N index (u within tile)
            const int k0 = (lane >> 4) << 1;        // K pair: lanes<16 -> {0,1}, >=16 -> {2,3}
            const float amask = (n == 0) ? 1.0f : 0.0f;  // only lanes 0/16 carry A row M=0
            v8f acc[4] = {};
            for (int kk = 0; kk < 32; ++kk) {
                const int r0 = r_base + kk*4;
                v2f a;
                a.x = amask * s_sim[r0 + k0];        // unconditional LDS broadcast + mask
                a.y = amask * s_sim[r0 + k0 + 1];
                const float* b0 = &s_assoc[(r0 + k0    )*U_ + n];
                const float* b1 = &s_assoc[(r0 + k0 + 1)*U_ + n];
                #pragma unroll
                for (int ut = 0; ut < 4; ++ut) {
                    v2f bb;
                    bb.x = b0[ut*16];
                    bb.y = b1[ut*16];
                    acc[ut] = __builtin_amdgcn_wmma_f32_16x16x4_f32(
                        false, a, false, bb, (short)0, acc[ut], false, false);
                }
            }
            if (lane < 16) {
                #pragma unroll
                for (int ut = 0; ut < 4; ++ut)
                    atomicAdd(&s_x[ut*16 + lane], acc[ut][0]);   // ds_add_f32
            }
        }
        __syncthreads();

        // ---- humble teacher grad + softmax output (threads 0..63) ----
        if (tid < U_) {
            const float xv = s_x[tid];
            s_gx[tid] = (tid == lab) ? fminf(xv - 1.0f, 0.0f)
                                     : fmaxf(xv + 1.0f, 0.0f);
        }
        __syncthreads();
        if (tid < U_) {
            float m = -__builtin_inff();
            for (int j = 0; j < U_; ++j) m = fmaxf(m, temp * s_x[j]);
            float sum = 0.0f;
            for (int j = 0; j < U_; ++j) sum += __expf(temp * s_x[j] - m);
            out[((size_t)b*T_ + t)*U_ + tid] = __expf(temp * s_x[tid] - m) / sum;
        }

        // ---- Phase C: g_s[r] = sum_u assoc[r][u]*g_x[u]  (WMMA, pre-update assoc) ----
        // A fragment (g_x) hoisted across the 8 r-tiles; 8 independent accumulators.
        {
            const int r_base = wave * 128;
            const int n  = lane & 15;               // N index (r within tile)
            const int k0 = (lane >> 4) << 1;        // K pair (u)
            const float amask = (n == 0) ? 1.0f : 0.0f;
            v8f acc[8] = {};
            for (int ku = 0; ku < 16; ++ku) {
                const int u0 = ku*4;
                v2f a;
                a.x = amask * s_gx[u0 + k0];
                a.y = amask * s_gx[u0 + k0 + 1];
                #pragma unroll
                for (int rt = 0; rt < 8; ++rt) {
                    const int r0 = r_base + rt*16;
                    v2f bb;
                    bb.x = s_assoc[(r0 + n)*U_ + u0 + k0];
                    bb.y = s_assoc[(r0 + n)*U_ + u0 + k0 + 1];
                    acc[rt] = __builtin_amdgcn_wmma_f32_16x16x4_f32(
                        false, a, false, bb, (short)0, acc[rt], false, false);
                }
            }
            if (lane < 16) {
                #pragma unroll
                for (int rt = 0; rt < 8; ++rt)
                    s_gs[r_base + rt*16 + lane] = acc[rt][0];
            }
        }
        __syncthreads();

        // ---- Phase D: g_att coefficient + rank-1 association update ----
        // dd/dS = (1/rho)*(S+eps)^(1/rho-1) = (1/rho)*d/(S+eps)  (no second pow)
        for (int k = 0; k < 4; ++k) {
            const int r = tid + k*256;
            const float gd = -beta * s_sim[r] * s_gs[r];
            s_coef[r] = gd * inv_rho * s_dd[r] / s_S[r];
        }
        for (int i = tid; i < R_*U_; i += 256) {
            const int r = i >> 6, u = i & 63;
            s_assoc[i] -= lrW * s_sim[r] * s_gx[u];
        }
        __syncthreads();

        // ---- Phase E: g_att[d] = sum_r coef[r]*diffpow[r,d]; update attention ----
        {
            const int d = tid & 63;
            const int stripe = tid >> 6;            // 4 r-stripes of 256
            const float zd = s_z[d];
            float acc = 0.0f;
            for (int r = stripe*256; r < stripe*256 + 256; ++r) {
                const float df = fabsf(zd - embed[(size_t)r*D_ + d]) + EPSF;
                acc += s_coef[r] * powf(df, rho);
            }
            s_red[tid] = acc;
        }
        __syncthreads();
        if (tid < D_) {
            const float g = s_red[tid] + s_red[tid+64] + s_red[tid+128] + s_red[tid+192];
            s_att[tid] = fmaxf(s_att[tid] - lrA * g, 0.0f);
        }
        __syncthreads();
    }
}

extern "C" void kernel_launch(void* const* d_in, const int* in_sizes, int n_in,
                              void* d_out, int out_size, void* d_ws, size_t ws_size,
                              hipStream_t stream) {
    (void)in_sizes; (void)n_in; (void)out_size; (void)d_ws; (void)ws_size;
    const int*   stim  = (const int*)  d_in[0];
    const int*   label = (const int*)  d_in[1];
    const float* embed = (const float*)d_in[2];
    const float* rho   = (const float*)d_in[3];
    const float* temp  = (const float*)d_in[4];
    const float* lrA   = (const float*)d_in[5];
    const float* lrW   = (const float*)d_in[6];
    const float* beta  = (const float*)d_in[7];
    float* out = (float*)d_out;

    // Allow >64KB dynamic LDS (CDNA5 WGP: 320KB). Host-side attr set, capture-safe.
    hipFuncSetAttribute(reinterpret_cast<const void*>(&alcove_scan_kernel),
                        hipFuncAttributeMaxDynamicSharedMemorySize, SMEM_BYTES);

    alcove_scan_kernel<<<B_, 256, SMEM_BYTES, stream>>>(
        stim, label, embed, rho, temp, lrA, lrW, beta, out);
}